// Sparsify2D_kactive_987842478201
// MI455X (gfx1250) — compile-verified
//
#include <hip/hip_runtime.h>
#include <stdint.h>
#include <stddef.h>

// ---------------------------------------------------------------------------
// Sparsify2D k-active: x [64, 384, 48, 48] f32. Per sample keep values
// >= (64th largest), else 0.  Exact 3-level radix select (12+12+8 bits) on a
// monotone uint mapping of the floats, then one streaming threshold pass.
// Memory-bound: ~453 MB compulsory traffic (~19.4 us @ 23.3 TB/s); extra
// selection reads mostly hit the 192 MB L2.
// ---------------------------------------------------------------------------

#define B_      64
#define CHW     (384 * 48 * 48)      // 884736 elements per sample
#define N4_     (CHW / 4)            // 221184 float4 per sample
#define KSEL    64
#define HBINS   4096
#define NCOPY   2                    // LDS histogram copies (bank-conflict relief)
#define TPB     256
#define HCHUNKS 64                   // blocks per sample, histogram passes
#define ACHUNKS 432                  // blocks per sample, apply pass

typedef float f4 __attribute__((ext_vector_type(4)));
typedef int   i4 __attribute__((ext_vector_type(4)));

#if defined(__AMDGCN__) && defined(__has_builtin)
#if __has_builtin(__builtin_amdgcn_global_load_async_to_lds_b128)
#define HAVE_ASYNC 1
#endif
#endif
#ifndef HAVE_ASYNC
#define HAVE_ASYNC 0
#endif

#if HAVE_ASYNC
typedef __attribute__((address_space(1))) i4* g_i4p;   // global int4*
typedef __attribute__((address_space(3))) i4* l_i4p;   // LDS int4*

__device__ __forceinline__ void async_copy_b128(const void* g, void* l) {
  // GLOBAL_LOAD_ASYNC_TO_LDS_B128: per-lane global -> LDS, tracked by ASYNCcnt
  __builtin_amdgcn_global_load_async_to_lds_b128((g_i4p)g, (l_i4p)l, 0, 0);
}
#endif

// Monotone map: larger float  <=>  larger unsigned.
__device__ __forceinline__ unsigned map_u(float f) {
  unsigned b = __float_as_uint(f);
  return (b & 0x80000000u) ? ~b : (b | 0x80000000u);
}

__global__ void __launch_bounds__(TPB) init_ws(unsigned* __restrict__ hist) {
  int i = (int)blockIdx.x * TPB + (int)threadIdx.x;
  if (i < B_ * HBINS) hist[i] = 0u;
}

// LEVEL 0: bin = u>>20 (all elements)
// LEVEL 1: bin = (u>>8)&0xFFF for elements with (u>>20)==prefix (12 bits)
// LEVEL 2: bin = u&0xFF      for elements with (u>>8) ==prefix (24 bits)
template <int LEVEL>
__global__ void __launch_bounds__(TPB) hist_pass(const float* __restrict__ x,
                                                 unsigned* __restrict__ ghist,
                                                 const unsigned* __restrict__ prefix) {
  __shared__ unsigned lh[NCOPY][HBINS];
#if HAVE_ASYNC
  __shared__ f4 stage[2][TPB];
#endif
  for (int i = threadIdx.x; i < NCOPY * HBINS; i += TPB) (&lh[0][0])[i] = 0u;
  __syncthreads();

  const int b = blockIdx.y;
  const unsigned pref = (LEVEL == 0) ? 0u : prefix[b];
  const f4* xb = (const f4*)(x + (size_t)b * CHW);
  unsigned* mh = lh[(threadIdx.x >> 7) & (NCOPY - 1)];
  const int stride = (int)gridDim.x * TPB;

  auto accum = [&](f4 v) {
#pragma unroll
    for (int c = 0; c < 4; ++c) {
      unsigned u = map_u(v[c]);
      if (LEVEL == 0) {
        atomicAdd(&mh[u >> 20], 1u);
      } else if (LEVEL == 1) {
        if ((u >> 20) == pref) atomicAdd(&mh[(u >> 8) & 0xFFFu], 1u);
      } else {
        if ((u >> 8) == pref) atomicAdd(&mh[u & 0xFFu], 1u);
      }
    }
  };

#if HAVE_ASYNC
  // Double-buffered async global->LDS pipeline (ASYNCcnt-tracked).
  int i = (int)blockIdx.x * TPB + (int)threadIdx.x;
  if (i < N4_) {
    async_copy_b128(&xb[i], &stage[0][threadIdx.x]);
    int buf = 0;
    for (;;) {
      const int inext = i + stride;
      if (inext < N4_) {
        async_copy_b128(&xb[inext], &stage[buf ^ 1][threadIdx.x]);
        asm volatile("s_wait_asynccnt 1" ::: "memory");
      } else {
        asm volatile("s_wait_asynccnt 0" ::: "memory");
      }
      accum(stage[buf][threadIdx.x]);
      if (inext >= N4_) break;
      i = inext;
      buf ^= 1;
    }
  }
#else
  for (int i = (int)blockIdx.x * TPB + (int)threadIdx.x; i < N4_; i += stride) {
    __builtin_prefetch(&xb[i + stride], 0, 0);  // global_prefetch_b8
    accum(xb[i]);
  }
#endif

  __syncthreads();
  unsigned* gh = ghist + (size_t)b * HBINS;
  for (int i2 = threadIdx.x; i2 < HBINS; i2 += TPB) {
    unsigned s = 0;
#pragma unroll
    for (int cpy = 0; cpy < NCOPY; ++cpy) s += lh[cpy][i2];
    if (s) atomicAdd(&gh[i2], s);
  }
}

// One block per sample: walk bins top-down to locate the bin holding the
// residual rank, update prefix / residual-k, zero the histogram for reuse.
template <int LEVEL>
__global__ void __launch_bounds__(TPB) scan_pass(unsigned* __restrict__ hist,
                                                 unsigned* __restrict__ prefix,
                                                 unsigned* __restrict__ krem,
                                                 float* __restrict__ thr) {
  const int b = blockIdx.x;
  unsigned* h = hist + (size_t)b * HBINS;
  if (threadIdx.x == 0) {
    const int bins = (LEVEL == 2) ? 256 : HBINS;
    unsigned k = (LEVEL == 0) ? (unsigned)KSEL : krem[b];
    unsigned cum = 0;
    unsigned sel = 0;
    for (int i = bins - 1; i >= 0; --i) {
      unsigned c = h[i];
      if (cum + c >= k) { sel = (unsigned)i; break; }
      cum += c;
    }
    krem[b] = k - cum;  // residual rank inside selected bin
    if (LEVEL == 0) {
      prefix[b] = sel;
    } else if (LEVEL == 1) {
      prefix[b] = (prefix[b] << 12) | sel;
    } else {
      const unsigned u = (prefix[b] << 8) | sel;  // exact k-th value (mapped)
      const unsigned fb = (u & 0x80000000u) ? (u ^ 0x80000000u) : ~u;
      thr[b] = __uint_as_float(fb);
    }
  }
  __syncthreads();
  for (int i = threadIdx.x; i < HBINS; i += TPB) h[i] = 0u;
}

__global__ void __launch_bounds__(TPB) apply_pass(const float* __restrict__ x,
                                                  float* __restrict__ out,
                                                  const float* __restrict__ thr) {
  const int b = blockIdx.y;
  const float t = thr[b];
  const f4* xb = (const f4*)(x + (size_t)b * CHW);
  f4* ob = (f4*)(out + (size_t)b * CHW);
  const int stride = (int)gridDim.x * TPB;
  for (int i = (int)blockIdx.x * TPB + (int)threadIdx.x; i < N4_; i += stride) {
    __builtin_prefetch(&xb[i + stride], 0, 0);       // global_prefetch_b8
    f4 v = __builtin_nontemporal_load(&xb[i]);        // last use of x
    f4 r;
#pragma unroll
    for (int c = 0; c < 4; ++c) r[c] = (v[c] >= t) ? v[c] : 0.0f;
    __builtin_nontemporal_store(r, &ob[i]);           // streaming store
  }
}

extern "C" void kernel_launch(void* const* d_in, const int* in_sizes, int n_in,
                              void* d_out, int out_size, void* d_ws, size_t ws_size,
                              hipStream_t stream) {
  (void)in_sizes; (void)n_in; (void)out_size; (void)ws_size;

  const float* x = (const float*)d_in[0];
  float* out = (float*)d_out;

  unsigned char* ws = (unsigned char*)d_ws;
  unsigned* hist   = (unsigned*)ws;                                   // 64*4096*4 = 1 MB
  unsigned* prefix = (unsigned*)(ws + (size_t)B_ * HBINS * sizeof(unsigned));
  unsigned* krem   = prefix + B_;
  float*    thr    = (float*)(krem + B_);

  init_ws<<<(B_ * HBINS + TPB - 1) / TPB, TPB, 0, stream>>>(hist);

  hist_pass<0><<<dim3(HCHUNKS, B_), TPB, 0, stream>>>(x, hist, prefix);
  scan_pass<0><<<B_, TPB, 0, stream>>>(hist, prefix, krem, thr);

  hist_pass<1><<<dim3(HCHUNKS, B_), TPB, 0, stream>>>(x, hist, prefix);
  scan_pass<1><<<B_, TPB, 0, stream>>>(hist, prefix, krem, thr);

  hist_pass<2><<<dim3(HCHUNKS, B_), TPB, 0, stream>>>(x, hist, prefix);
  scan_pass<2><<<B_, TPB, 0, stream>>>(hist, prefix, krem, thr);

  apply_pass<<<dim3(ACHUNKS, B_), TPB, 0, stream>>>(x, out, thr);
}